// Attention_85813446574600
// MI455X (gfx1250) — compile-verified
//
#include <hip/hip_runtime.h>
#include <hip/hip_bf16.h>

typedef __attribute__((ext_vector_type(2))) float v2f;
typedef __attribute__((ext_vector_type(8))) float v8f;

constexpr int Bv = 32, Tv = 512, Cv = 256, Fv = 512;
constexpr int CHUNKS = 8;            // split T 8 ways for memory-level parallelism
constexpr int TPC    = Tv / CHUNKS;  // 64 time rows per chunk

// ---------------------------------------------------------------------------
// Stage 1: partial column sums via V_WMMA_F32_16X16X4_F32.
// A = all-ones 16x4  =>  D[m,n] = sum_k B[k,n]  (robust to K-permutation in the
// B VGPR layout; only lane%16 -> column N matters).
// Each wave owns (b, 16-wide f-tile, 64-row T-chunk): 16 chained WMMAs.
// Partial sums are staged into C-rows 0..7 of d_out: out[b, chunk, f].
// ---------------------------------------------------------------------------
__global__ void __launch_bounds__(256) wmma_partial_sums(
    const float* __restrict__ x, float* __restrict__ out) {
  const int wave  = blockIdx.x * 8 + (threadIdx.x >> 5);
  const int lane  = threadIdx.x & 31;
  const int b     = wave >> 8;        // 256 waves per batch (32 f-tiles * 8 chunks)
  const int rem   = wave & 255;
  const int ftile = rem >> 3;         // 32 tiles of 16 columns
  const int chunk = rem & 7;
  const int n     = lane & 15;        // column within tile (N = lane % 16)
  const int khalf = (lane >> 4) << 1; // K offset per half-wave: 0 or 2

  const float* xb = x + (size_t)b * Tv * Fv + (size_t)ftile * 16 + n;
  const int t0 = chunk * TPC;

  v2f a; a[0] = 1.0f; a[1] = 1.0f;    // all-ones A matrix
  v8f acc = {};
#pragma unroll
  for (int t = 0; t < TPC; t += 4) {
    v2f bm;
    bm[0] = xb[(size_t)(t0 + t + khalf) * Fv];
    bm[1] = xb[(size_t)(t0 + t + khalf + 1) * Fv];
    // 8 args: (neg_a, A, neg_b, B, c_mod, C, reuse_a, reuse_b)
    acc = __builtin_amdgcn_wmma_f32_16x16x4_f32(false, a, false, bm,
                                                (short)0, acc, false, false);
  }
  // Every D row holds the same column sum; lanes 0..15 cover N = 0..15.
  if (lane < 16) {
    out[(size_t)b * Cv * Fv + (size_t)chunk * Fv + (size_t)ftile * 16 + n] = acc[0];
  }
}

// ---------------------------------------------------------------------------
// Stage 1.5: total[b,f] = sum over the 8 chunk partials; stored into C-row 8.
// ---------------------------------------------------------------------------
__global__ void __launch_bounds__(256) reduce_chunks(float* __restrict__ out) {
  const int i = blockIdx.x * 256 + threadIdx.x; // B*F = 16384 threads
  const int b = i >> 9;
  const int f = i & 511;
  const float* p = out + (size_t)b * Cv * Fv + f;
  float s = 0.0f;
#pragma unroll
  for (int c = 0; c < CHUNKS; ++c) s += p[(size_t)c * Fv];
  out[(size_t)b * Cv * Fv + (size_t)CHUNKS * Fv + f] = s;
}

// ---------------------------------------------------------------------------
// Stage 2: broadcast row 8 to all 256 C rows with coalesced float4 stores.
// (c==8 rewrites the identical value it reads -> deterministic.)
// ---------------------------------------------------------------------------
__global__ void __launch_bounds__(256) broadcast_rows(float* __restrict__ out) {
  const int i  = blockIdx.x * 256 + threadIdx.x; // B*C*F/4 = 1,048,576 threads
  const int b  = i >> 15;                        // C*F/4 = 32768 per batch
  const int r  = i & 32767;
  const int c  = r >> 7;                         // F/4 = 128 float4 per row
  const int f4 = r & 127;
  const float4* src =
      (const float4*)(out + (size_t)b * Cv * Fv + (size_t)CHUNKS * Fv);
  float4 v = src[f4];
  float4* dst = (float4*)(out + (size_t)b * Cv * Fv + (size_t)c * Fv);
  dst[f4] = v;
}

extern "C" void kernel_launch(void* const* d_in, const int* in_sizes, int n_in,
                              void* d_out, int out_size, void* d_ws, size_t ws_size,
                              hipStream_t stream) {
  (void)in_sizes; (void)n_in; (void)d_ws; (void)ws_size; (void)out_size;
  const float* x = (const float*)d_in[0];  // [B,T,F] f32; context/W/b cancel out
  float* out = (float*)d_out;              // [B,C,F] f32

  // Stage 1: 8192 waves -> 1024 blocks of 8 waves.
  wmma_partial_sums<<<1024, 256, 0, stream>>>(x, out);
  // Stage 1.5: 16384 threads.
  reduce_chunks<<<64, 256, 0, stream>>>(out);
  // Stage 2: 1,048,576 threads.
  broadcast_rows<<<4096, 256, 0, stream>>>(out);
}